// MyModel_61933428414756
// MI455X (gfx1250) — compile-verified
//
#include <hip/hip_runtime.h>
#include <hip/hip_bf16.h>

typedef __attribute__((ext_vector_type(2))) float v2f;
typedef __attribute__((ext_vector_type(8))) float v8f;

#define THETA  1.461661507209034f
#define WAVES_PER_BLOCK 2
#define MATS_PER_WAVE_LDS 5120   // A, A2, A3, A4, P  (5 x 1024 floats)

// ---------------------------------------------------------------------------
// One wave32 per 32x32 fp32 matrix; matmuls via V_WMMA_F32_16X16X4_F32.
// expm via degree-15 Taylor evaluated Paterson-Stockmeyer style (6 matmuls)
// followed by s repeated squarings.
//
// Fragment layouts (CDNA5 ISA 7.12.2):
//   A 16x4 (MxK):  lanes 0-15 hold M=lane,    VGPR0=K0,   VGPR1=K0+1
//                  lanes 16-31 hold M=lane-16, VGPR0=K0+2, VGPR1=K0+3
//   B 4x16 (KxN):  VGPR0: lanes0-15 row K0   (N=lane), lanes16-31 row K0+2
//                  VGPR1: lanes0-15 row K0+1,          lanes16-31 row K0+3
//   C 16x16:       VGPR r: lanes0-15 M=r, lanes16-31 M=r+8, N=lane&15
// ---------------------------------------------------------------------------

#define LDS_FENCE() asm volatile("s_wait_dscnt 0" ::: "memory")

// A-fragments of a 32x32 matrix -> 32 VGPRs: af[k-step][tile-row]
__device__ __forceinline__ void load_afrags(const float* Am, v2f af[8][2],
                                            int lane) {
  const int mlo = lane & 15;
  const int hi  = (lane >> 4) & 1;
#pragma unroll
  for (int ks = 0; ks < 8; ++ks) {
    const int kk = ks * 4 + hi * 2;
#pragma unroll
    for (int ti = 0; ti < 2; ++ti)
      af[ks][ti] = *(const v2f*)&Am[(ti * 16 + mlo) * 32 + kk];  // ds_load_b64
  }
}

__device__ __forceinline__ void zero_acc(v8f acc[2][2]) {
  v8f z = {};
  acc[0][0] = z; acc[0][1] = z; acc[1][0] = z; acc[1][1] = z;
}

// acc init from an LDS matrix in C-tile layout (WMMA accumulator operand)
__device__ __forceinline__ void load_ctiles(const float* Bm, v8f acc[2][2],
                                            int lane) {
  const int mlo = lane & 15;
  const int hi  = (lane >> 4) & 1;
#pragma unroll
  for (int ti = 0; ti < 2; ++ti)
#pragma unroll
    for (int tj = 0; tj < 2; ++tj)
#pragma unroll
      for (int r = 0; r < 8; ++r)
        acc[ti][tj][r] = Bm[(ti * 16 + hi * 8 + r) * 32 + tj * 16 + mlo];
}

// acc += Afrags * B(LDS)
__device__ __forceinline__ void matmul_acc(const v2f af[8][2], const float* Bm,
                                           v8f acc[2][2], int lane) {
  const int mlo = lane & 15;
  const int hi  = (lane >> 4) & 1;
#pragma unroll
  for (int ks = 0; ks < 8; ++ks) {
    const int kk = ks * 4 + hi * 2;
    v2f b[2];
#pragma unroll
    for (int tj = 0; tj < 2; ++tj) {
      const int n = tj * 16 + mlo;
      b[tj].x = Bm[kk * 32 + n];
      b[tj].y = Bm[(kk + 1) * 32 + n];
    }
#pragma unroll
    for (int ti = 0; ti < 2; ++ti)
#pragma unroll
      for (int tj = 0; tj < 2; ++tj)
        acc[ti][tj] = __builtin_amdgcn_wmma_f32_16x16x4_f32(
            /*neg_a=*/false, af[ks][ti], /*neg_b=*/false, b[tj],
            /*c_mod=*/(short)0, acc[ti][tj],
            /*reuse_a=*/false, /*reuse_b=*/false);
  }
}

__device__ __forceinline__ void store_ctiles(float* Dm, const v8f acc[2][2],
                                             int lane) {
  const int mlo = lane & 15;
  const int hi  = (lane >> 4) & 1;
#pragma unroll
  for (int ti = 0; ti < 2; ++ti)
#pragma unroll
    for (int tj = 0; tj < 2; ++tj)
#pragma unroll
      for (int r = 0; r < 8; ++r)
        Dm[(ti * 16 + hi * 8 + r) * 32 + tj * 16 + mlo] = acc[ti][tj][r];
}

// One Paterson-Stockmeyer step:  P = P*A4 + Binit
__device__ __forceinline__ void ps_step(float* P, const float* A4m,
                                        const float* Binit, int lane) {
  v2f pf[8][2];
  load_afrags(P, pf, lane);
  v8f acc[2][2];
  load_ctiles(Binit, acc, lane);
  matmul_acc(pf, A4m, acc, lane);
  store_ctiles(P, acc, lane);
  LDS_FENCE();
}

__launch_bounds__(WAVES_PER_BLOCK * 32)
__global__ void expm32_wmma(const float* __restrict__ x,
                            float* __restrict__ y, int nmat) {
  __shared__ float lds[WAVES_PER_BLOCK * MATS_PER_WAVE_LDS];  // 40 KB

  const int lane = threadIdx.x & 31;
  const int wave = threadIdx.x >> 5;
  const int mat  = blockIdx.x * WAVES_PER_BLOCK + wave;
  if (mat >= nmat) return;                  // wave-uniform; EXEC stays all-1s

  float* A   = &lds[wave * MATS_PER_WAVE_LDS];
  float* A2  = A  + 1024;
  float* A3  = A2 + 1024;
  float* A4m = A3 + 1024;
  float* P   = A4m + 1024;

  const float4* src4 = (const float4*)(x + (size_t)mat * 1024);
  float4*       dst4 = (float4*)(y + (size_t)mat * 1024);
  float4*       Av   = (float4*)A;
  float4*       A2v  = (float4*)A2;
  float4*       A3v  = (float4*)A3;
  float4*       Pv   = (float4*)P;

  // ---- load matrix into LDS (coalesced b128) ----
#pragma unroll
  for (int c = 0; c < 8; ++c) Av[c * 32 + lane] = src4[c * 32 + lane];
  LDS_FENCE();

  // ---- induced 1-norm: lane j sums |column j|, then max over lanes ----
  float colsum = 0.0f;
#pragma unroll
  for (int i = 0; i < 32; ++i) colsum += fabsf(A[i * 32 + lane]);
#pragma unroll
  for (int off = 16; off > 0; off >>= 1)
    colsum = fmaxf(colsum, __shfl_xor(colsum, off, 32));

  float sf = floorf(log2f(colsum / THETA)) + 1.0f;
  sf = fmaxf(sf, 0.0f);
  const int   s     = __builtin_amdgcn_readfirstlane((int)sf);
  const float scale = exp2f(-sf);

  // ---- A *= 2^-s  (lane owns row `lane`) ----
#pragma unroll
  for (int c = 0; c < 8; ++c) {
    float4 v = Av[lane * 8 + c];
    v.x *= scale; v.y *= scale; v.z *= scale; v.w *= scale;
    Av[lane * 8 + c] = v;
  }
  LDS_FENCE();

  // ---- powers: A2 = A*A, A3 = A*A2, A4 = A2*A2 ----
  {
    v2f af[8][2];
    load_afrags(A, af, lane);
    v8f acc[2][2];
    zero_acc(acc);
    matmul_acc(af, A, acc, lane);
    store_ctiles(A2, acc, lane);
    LDS_FENCE();
    zero_acc(acc);
    matmul_acc(af, A2, acc, lane);
    store_ctiles(A3, acc, lane);
    LDS_FENCE();
  }
  {
    v2f a2f[8][2];
    load_afrags(A2, a2f, lane);
    v8f acc[2][2];
    zero_acc(acc);
    matmul_acc(a2f, A2, acc, lane);
    store_ctiles(A4m, acc, lane);
    LDS_FENCE();
  }

  // ---- elementwise: Bj = c[4j]*I + c[4j+1]*A + c[4j+2]*A2 + c[4j+3]*A3 ----
  // B0 -> A slot, B1 -> A2 slot, B2 -> A3 slot, B3 -> P slot.
  {
    const float C0[4] = {1.0f, 1.0f / 24.0f, 1.0f / 40320.0f,
                         1.0f / 479001600.0f};
    const float C1[4] = {1.0f, 1.0f / 120.0f, 1.0f / 362880.0f,
                         1.0f / 6227020800.0f};
    const float C2[4] = {0.5f, 1.0f / 720.0f, 1.0f / 3628800.0f,
                         1.0f / 87178291200.0f};
    const float C3[4] = {1.0f / 6.0f, 1.0f / 5040.0f, 1.0f / 39916800.0f,
                         1.0f / 1307674368000.0f};
#pragma unroll
    for (int c = 0; c < 8; ++c) {
      const float4 a  = Av[lane * 8 + c];
      const float4 a2 = A2v[lane * 8 + c];
      const float4 a3 = A3v[lane * 8 + c];
      float4 b[4];
#pragma unroll
      for (int j = 0; j < 4; ++j) {
        b[j].x = fmaf(C3[j], a3.x, fmaf(C2[j], a2.x, C1[j] * a.x));
        b[j].y = fmaf(C3[j], a3.y, fmaf(C2[j], a2.y, C1[j] * a.y));
        b[j].z = fmaf(C3[j], a3.z, fmaf(C2[j], a2.z, C1[j] * a.z));
        b[j].w = fmaf(C3[j], a3.w, fmaf(C2[j], a2.w, C1[j] * a.w));
        if ((lane >> 2) == c) (&b[j].x)[lane & 3] += C0[j];   // diagonal
      }
      Av[lane * 8 + c]  = b[0];
      A2v[lane * 8 + c] = b[1];
      A3v[lane * 8 + c] = b[2];
      Pv[lane * 8 + c]  = b[3];
    }
    LDS_FENCE();
  }

  // ---- Paterson-Stockmeyer: P = ((B3*A4 + B2)*A4 + B1)*A4 + B0 ----
  ps_step(P, A4m, A3, lane);   // P = B3*A4 + B2
  ps_step(P, A4m, A2, lane);   // P = P*A4  + B1
  ps_step(P, A4m, A,  lane);   // P = P*A4  + B0   -> expm(A/2^s)

  // ---- repeated squaring: P = P*P, s times ----
#pragma unroll 1
  for (int it = 0; it < s; ++it) {
    v2f pf[8][2];
    load_afrags(P, pf, lane);
    v8f acc[2][2];
    zero_acc(acc);
    matmul_acc(pf, P, acc, lane);
    store_ctiles(P, acc, lane);
    LDS_FENCE();
  }

  // ---- write result (coalesced b128) ----
#pragma unroll
  for (int c = 0; c < 8; ++c) dst4[c * 32 + lane] = Pv[c * 32 + lane];
}

extern "C" void kernel_launch(void* const* d_in, const int* in_sizes, int n_in,
                              void* d_out, int out_size, void* d_ws, size_t ws_size,
                              hipStream_t stream) {
  const float* x = (const float*)d_in[0];
  float*       y = (float*)d_out;
  const int nmat = in_sizes[0] / 1024;          // 65536 matrices of 32x32
  const int blocks = (nmat + WAVES_PER_BLOCK - 1) / WAVES_PER_BLOCK;
  hipLaunchKernelGGL(expm32_wmma, dim3(blocks), dim3(WAVES_PER_BLOCK * 32), 0,
                     stream, x, y, nmat);
}